// VSTM_78503412236842
// MI455X (gfx1250) — compile-verified
//
#include <hip/hip_runtime.h>
#include <hip/hip_bf16.h>
#include <math.h>

typedef __attribute__((ext_vector_type(2))) float v2f;
typedef __attribute__((ext_vector_type(8))) float v8f;

#define B_SZ    64
#define IN_CH   32
#define OUT_CH  32
#define N_MAPS  64
#define HW      4096
#define CTRL_N  512
#define NGATE   4160   // NF + NI + NO
#define NI_OFF  64     // start of input gates in ctrl row
#define NO_OFF  2112   // start of output gates in ctrl row
#define NT      128    // spatial columns per workgroup tile

#if defined(__AMDGCN__) && \
    __has_builtin(__builtin_amdgcn_global_load_async_to_lds_b128) && \
    __has_builtin(__builtin_amdgcn_s_wait_asynccnt)
#define USE_ASYNC_LDS 1
// builtin signature (from hipcc diagnostic): param0 = AS1 (global) int4-vector ptr,
// param1 = AS3 (LDS) int4-vector ptr, then two constant ints (offset, cpol).
typedef int b128_t __attribute__((vector_size(16)));
#define GLOBAL_B128(p) ((__attribute__((address_space(1))) b128_t*)(p))
#define LDS_B128(p)    ((__attribute__((address_space(3))) b128_t*)(p))
#else
#define USE_ASYNC_LDS 0
#endif

// -------------------------------------------------------------------------
// Kernel 1: ctrl(64 x 4160) = controls(64 x 512) @ W(4160 x 512)^T
// One wave per 16x16 output tile, V_WMMA_F32_16X16X4_F32, K-loop of 128.
// A fragment: lanes 0-15 hold row m0+lane K={kk,kk+1}; lanes 16-31 K={kk+2,kk+3}.
// B fragment mirrors with N = n0 + (lane&15); both are contiguous float2 loads.
// W / controls stay regular-temporal: W is re-read by the 4 M-tiles via L2.
// -------------------------------------------------------------------------
__global__ __launch_bounds__(32)
void vstm_gates(const float* __restrict__ controls,   // (64, 512)
                const float* __restrict__ Wc,         // (4160, 512)
                float* __restrict__ ctrl)             // (64, 4160)
{
    const int ntile = blockIdx.x;          // 0..259
    const int mtile = blockIdx.y;          // 0..3
    const int lane  = threadIdx.x;
    const int lr    = lane & 15;
    const int half  = lane >> 4;
    const int koff  = half * 2;

    const int m0 = mtile * 16;
    const int n0 = ntile * 16;

    const float* arow = controls + (size_t)(m0 + lr) * CTRL_N;
    const float* brow = Wc       + (size_t)(n0 + lr) * CTRL_N;

    v8f acc = {};
    for (int kk = 0; kk < CTRL_N; kk += 4) {
        v2f a, b;
        a.x = arow[kk + koff];
        a.y = arow[kk + koff + 1];
        b.x = brow[kk + koff];
        b.y = brow[kk + koff + 1];
        acc = __builtin_amdgcn_wmma_f32_16x16x4_f32(false, a, false, b,
                                                    (short)0, acc, false, false);
    }
#pragma unroll
    for (int i = 0; i < 8; ++i) {
        int m = m0 + i + half * 8;             // C layout: VGPR i -> M=i / M=8+i
        ctrl[(size_t)m * NGATE + n0 + lr] = acc[i];
    }
}

// -------------------------------------------------------------------------
// Kernel 2: fused  gated = in_gates @ inputs ; ns = gated + state*forget ;
//                  th = tanh(ns) ; outputs = out_gates @ th
// One block per (batch, 128-column spatial tile). 8 waves.
// GEMM1: M=64 (maps) x K=32 (in_ch) x N=128   -> 32 wave-tiles, 4 per wave
// GEMM2: M=32 (out)  x K=64 (maps)  x N=128   -> 16 wave-tiles, 2 per wave
// Streaming traffic (inputs/state/new_state/outputs) is touch-once -> NT hints.
// -------------------------------------------------------------------------
__global__ __launch_bounds__(256)
void vstm_fused(const float* __restrict__ inputs,    // (B, 32, HW)
                const float* __restrict__ state,     // (B, 64, HW)
                const float* __restrict__ ctrl,      // (B, 4160)
                float* __restrict__ outputs,         // (B, 32, HW)
                float* __restrict__ new_state)       // (B, 64, HW)
{
    __shared__ float in_lds[IN_CH * NT];     // 16 KB: inputs tile [c][n]
    __shared__ float th_lds[N_MAPS * NT];    // 32 KB: tanh(new_state) tile [m][n]
    __shared__ float fg_lds[N_MAPS];         // forget gates

    const int b    = blockIdx.y;
    const int hw0  = blockIdx.x * NT;
    const int tid  = threadIdx.x;
    const int lane = tid & 31;
    const int wave = tid >> 5;               // 0..7
    const int lr   = lane & 15;
    const int half = lane >> 4;
    const int koff = half * 2;

    const float* cb = ctrl + (size_t)b * NGATE;

    // warm the caches with this block's state rows (consumed in GEMM1 epilogue)
    if (tid < 2 * N_MAPS) {
        int m = tid >> 1;
        int o = (tid & 1) * (NT / 2);
        __builtin_prefetch(&state[((size_t)b * N_MAPS + m) * HW + hw0 + o], 0, 1);
    }

    // ---- stage inputs tile + forget gates into LDS ----
#if USE_ASYNC_LDS
    for (int q = tid; q < (IN_CH * NT) / 4; q += 256) {
        int c = q / (NT / 4), n4 = (q - c * (NT / 4)) * 4;
        const float* gsrc = inputs + ((size_t)b * IN_CH + c) * HW + hw0 + n4;
        __builtin_amdgcn_global_load_async_to_lds_b128(
            GLOBAL_B128(gsrc), LDS_B128(&in_lds[c * NT + n4]), 0, 0);
    }
    __builtin_amdgcn_s_wait_asynccnt(0);
#else
    for (int idx = tid; idx < IN_CH * NT; idx += 256) {
        int c = idx / NT, n = idx - c * NT;
        in_lds[idx] =
            __builtin_nontemporal_load(&inputs[((size_t)b * IN_CH + c) * HW + hw0 + n]);
    }
#endif
    if (tid < N_MAPS) fg_lds[tid] = cb[tid];
    __syncthreads();

    // ---- GEMM1: gated[m][n] = sum_c in_gates[m][c] * in_lds[c][n] ----
    {
        const int m0 = (wave & 3) * 16;            // map tile
        const int ng = (wave >> 2) * 4;            // first of 4 N-tiles
        const float* ga = cb + NI_OFF + (size_t)(m0 + lr) * IN_CH;

        v8f acc[4] = {{}, {}, {}, {}};
        for (int kk = 0; kk < IN_CH; kk += 4) {
            v2f a;
            a.x = ga[kk + koff];
            a.y = ga[kk + koff + 1];
#pragma unroll
            for (int t = 0; t < 4; ++t) {
                int n0 = (ng + t) * 16;
                v2f bf;
                bf.x = in_lds[(kk + koff)     * NT + n0 + lr];
                bf.y = in_lds[(kk + koff + 1) * NT + n0 + lr];
                acc[t] = __builtin_amdgcn_wmma_f32_16x16x4_f32(
                    false, a, false, bf, (short)0, acc[t], false, false);
            }
        }

        // epilogue: ns = gated + state*forget ; store ns (NT) ; tanh -> LDS
#pragma unroll
        for (int t = 0; t < 4; ++t) {
            int n = (ng + t) * 16 + lr;
#pragma unroll
            for (int i = 0; i < 8; ++i) {
                int m = m0 + i + half * 8;
                size_t gidx = ((size_t)b * N_MAPS + m) * HW + hw0 + n;
                float sv = __builtin_nontemporal_load(&state[gidx]);
                float ns = acc[t][i] + sv * fg_lds[m];
                __builtin_nontemporal_store(ns, &new_state[gidx]);
                th_lds[m * NT + n] = tanhf(ns);
            }
        }
    }
    __syncthreads();

    // ---- GEMM2: outputs[o][n] = sum_m out_gates[o][m] * th_lds[m][n] ----
    {
        const int o0  = (wave & 1) * 16;           // out-channel tile
        const int ng2 = (wave >> 1) * 2;           // first of 2 N-tiles
        const float* go = cb + NO_OFF + (size_t)(o0 + lr) * N_MAPS;

        v8f acc[2] = {{}, {}};
        for (int kk = 0; kk < N_MAPS; kk += 4) {
            v2f a;
            a.x = go[kk + koff];
            a.y = go[kk + koff + 1];
#pragma unroll
            for (int t = 0; t < 2; ++t) {
                int n0 = (ng2 + t) * 16;
                v2f bf;
                bf.x = th_lds[(kk + koff)     * NT + n0 + lr];
                bf.y = th_lds[(kk + koff + 1) * NT + n0 + lr];
                acc[t] = __builtin_amdgcn_wmma_f32_16x16x4_f32(
                    false, a, false, bf, (short)0, acc[t], false, false);
            }
        }
#pragma unroll
        for (int t = 0; t < 2; ++t) {
            int n = (ng2 + t) * 16 + lr;
#pragma unroll
            for (int i = 0; i < 8; ++i) {
                int o = o0 + i + half * 8;
                __builtin_nontemporal_store(
                    acc[t][i], &outputs[((size_t)b * OUT_CH + o) * HW + hw0 + n]);
            }
        }
    }
}

// -------------------------------------------------------------------------
extern "C" void kernel_launch(void* const* d_in, const int* in_sizes, int n_in,
                              void* d_out, int out_size, void* d_ws, size_t ws_size,
                              hipStream_t stream) {
    const float* inputs   = (const float*)d_in[0];  // (64,32,64,64)
    const float* state    = (const float*)d_in[1];  // (64,64,64,64)
    const float* controls = (const float*)d_in[2];  // (64,512)
    const float* Wc       = (const float*)d_in[3];  // (4160,512)

    float* out_outputs   = (float*)d_out;                               // (64,32,64,64)
    float* out_new_state = (float*)d_out + (size_t)B_SZ * OUT_CH * HW;  // (64,64,64,64)
    float* ctrl_ws       = (float*)d_ws;                                // (64,4160) ~1.04 MB

    dim3 g1(NGATE / 16, B_SZ / 16);          // 260 x 4 tiles
    vstm_gates<<<g1, 32, 0, stream>>>(controls, Wc, ctrl_ws);

    dim3 g2(HW / NT, B_SZ);                  // 32 x 64 blocks
    vstm_fused<<<g2, 256, 0, stream>>>(inputs, state, ctrl_ws,
                                       out_outputs, out_new_state);
}